// MultiHeadSpMM_20968030339290
// MI455X (gfx1250) — compile-verified
//
#include <hip/hip_runtime.h>

// Multi-head SpMM via WMMA segmented reduction on gfx1250 (wave32).
// out[r, :] = sum_{e: row[e]==r} attention[e, hd] * h[col[e], hd, :]
// N=100000, E=1.6M, H=4, D=8, H*D=32 (== wave width).
//
// Per 4 edges: one-hot selector A (16x4) x message matrix B (4x16) accumulated
// into two 16x16 f32 C tiles via V_WMMA_F32_16X16X4_F32. C tile 0 holds even
// dims {2n}, tile 1 odd dims {2n+1}, so each lane's two h elements per edge are
// one coalesced float2 load and share a single attention scalar.

typedef __attribute__((ext_vector_type(2))) float v2f;
typedef __attribute__((ext_vector_type(8))) float v8f;
typedef __attribute__((__vector_size__(4 * sizeof(int)))) int v4i;

#define SPMM_H  4
#define SPMM_HD 32
#define CHUNK   256   // edges per wave (one wave per block)

#if __has_builtin(__builtin_amdgcn_global_load_async_to_lds_b128)
#define HAVE_ASYNC_LDS 1
#else
#define HAVE_ASYNC_LDS 0
#endif

__global__ void zero_f4(float4* __restrict__ out, int n4) {
  int i = blockIdx.x * blockDim.x + threadIdx.x;
  if (i < n4) out[i] = make_float4(0.f, 0.f, 0.f, 0.f);
}

// Fallback: one edge, all 32 lanes cover the 32 (head,dim) outputs.
__device__ __forceinline__ void edge_atomic(const int* __restrict__ row,
                                            const int* __restrict__ col,
                                            const float* __restrict__ att,
                                            const float* __restrict__ h,
                                            float* __restrict__ out,
                                            long e, int lane) {
  int   hd = lane >> 3;
  float a  = att[e * SPMM_H + hd];
  float x  = h[(size_t)col[e] * SPMM_HD + lane];
  atomicAdd(out + (size_t)row[e] * SPMM_HD + lane, a * x);
}

// Flush the 16-row x 32-dim accumulator window. c0 holds dims {2n}, c1 {2n+1}.
__device__ __forceinline__ void flush_tile(float* __restrict__ out, int base_row,
                                           int lane, v8f& c0, v8f& c1, int N) {
  const int n16 = lane & 15;
  const int hi  = lane >> 4;
  const int d0  = 2 * n16;
#pragma unroll
  for (int j = 0; j < 8; ++j) {
    int r = base_row + j + hi * 8;            // C layout: VGPR j -> M = j + 8*hi
    if (r < N) {
      float v0 = c0[j], v1 = c1[j];
      if (v0 != 0.f) atomicAdd(out + (size_t)r * SPMM_HD + d0, v0);
      if (v1 != 0.f) atomicAdd(out + (size_t)r * SPMM_HD + d0 + 1, v1);
    }
  }
  v8f z = {};
  c0 = z;
  c1 = z;
}

__global__ __launch_bounds__(32)
void spmm_wmma(const int* __restrict__ row, const int* __restrict__ col,
               const float* __restrict__ att, const float* __restrict__ h,
               float* __restrict__ out, int E, int N) {
  __shared__ alignas(16) int   scol[CHUNK];
  __shared__ alignas(16) float satt[CHUNK * SPMM_H];

  const int lane = threadIdx.x;          // 0..31, one wave per block
  const int n16  = lane & 15;            // tile N index / A-matrix M index
  const int hi   = lane >> 4;
  const int k0   = hi ? 2 : 0;           // K slots this lane holds (VGPR0, VGPR1)
  const int hh   = n16 >> 2;             // head for dims {2*n16, 2*n16+1}

  long eBeg = (long)blockIdx.x * CHUNK;
  long eEnd = eBeg + CHUNK; if (eEnd > E) eEnd = E;
  if (eBeg >= eEnd) return;

  if (eEnd - eBeg < CHUNK) {             // partial tail chunk: simple path
    for (long t = eBeg; t < eEnd; ++t) edge_atomic(row, col, att, h, out, t, lane);
    return;
  }

  // ---- Stage this chunk's col (1KB) + attention (4KB) into LDS ----
#if HAVE_ASYNC_LDS
  {
    const int*   gcol = col + eBeg;
    const float* gatt = att + eBeg * SPMM_H;
#pragma unroll
    for (int p = 0; p < CHUNK / 128; ++p)            // 128 ints per pass
      __builtin_amdgcn_global_load_async_to_lds_b128(
          (v4i*)(gcol + p * 128 + 4 * lane),
          (v4i*)(scol + p * 128 + 4 * lane), 0, 0);
#pragma unroll
    for (int p = 0; p < CHUNK * SPMM_H / 128; ++p)   // 128 floats per pass
      __builtin_amdgcn_global_load_async_to_lds_b128(
          (v4i*)(gatt + p * 128 + 4 * lane),
          (v4i*)(satt + p * 128 + 4 * lane), 0, 0);
#if __has_builtin(__builtin_amdgcn_s_wait_asynccnt)
    __builtin_amdgcn_s_wait_asynccnt(0);
#else
    asm volatile("s_wait_asynccnt 0x0" ::: "memory");
#endif
  }
#else
  for (int t = lane; t < CHUNK; t += 32)          scol[t] = col[eBeg + t];
  for (int t = lane; t < CHUNK * SPMM_H; t += 32) satt[t] = att[eBeg * SPMM_H + t];
#endif
  __syncthreads();

  v8f c0 = {}, c1 = {};
  int base_row = row[eBeg];

  for (long e = eBeg; e < eEnd; e += 4) {
    __builtin_prefetch(row + e + 64, 0, 0);        // row stream ahead

    const int g = (int)(e - eBeg);
    int r0g = row[e];                              // uniform -> s_load
    int r3g = row[e + 3];                          // rows sorted: group min/max

    if (r3g - r0g >= 16) {                         // rare wide span
      for (int t = 0; t < 4; ++t) edge_atomic(row, col, att, h, out, e + t, lane);
      continue;
    }
    if (r3g >= base_row + 16) {                    // slide 16-row window
      flush_tile(out, base_row, lane, c0, c1, N);
      base_row = r0g;
    }

    // The two edges (K slots k0, k0+1) this lane feeds: paired b64 loads.
    int2  rr = *(const int2*)(row + e + k0);
    int2  cc = *(const int2*)(scol + g + k0);
    float aa = satt[(g + k0) * SPMM_H + hh];
    float ab = satt[(g + k0 + 1) * SPMM_H + hh];

    // A: one-hot selector S[m][k] = (row[e_k] - base_row == m)
    v2f A;
    A.x = (rr.x - base_row == n16) ? 1.0f : 0.0f;
    A.y = (rr.y - base_row == n16) ? 1.0f : 0.0f;

    // B: messages, dims {2*n16, 2*n16+1} -> one float2 gather per edge
    float2 ha = *(const float2*)(h + (size_t)cc.x * SPMM_HD + 2 * n16);
    float2 hb = *(const float2*)(h + (size_t)cc.y * SPMM_HD + 2 * n16);

    v2f B0, B1;
    B0.x = aa * ha.x;  B0.y = ab * hb.x;           // even dims
    B1.x = aa * ha.y;  B1.y = ab * hb.y;           // odd dims

    c0 = __builtin_amdgcn_wmma_f32_16x16x4_f32(false, A, false, B0, (short)0, c0, false, false);
    c1 = __builtin_amdgcn_wmma_f32_16x16x4_f32(false, A, false, B1, (short)0, c1, false, false);
  }

  flush_tile(out, base_row, lane, c0, c1, N);
}

extern "C" void kernel_launch(void* const* d_in, const int* in_sizes, int n_in,
                              void* d_out, int out_size, void* d_ws, size_t ws_size,
                              hipStream_t stream) {
  const int*   row = (const int*)d_in[0];
  const int*   col = (const int*)d_in[1];
  const float* att = (const float*)d_in[2];
  const float* h   = (const float*)d_in[3];
  float*       out = (float*)d_out;

  int E = in_sizes[0];
  int N = in_sizes[3] / SPMM_HD;         // h holds N*H*D floats

  int n4 = out_size / 4;                 // N*32 divisible by 4
  zero_f4<<<(n4 + 255) / 256, 256, 0, stream>>>((float4*)d_out, n4);

  int blocks = (E + CHUNK - 1) / CHUNK;
  spmm_wmma<<<blocks, 32, 0, stream>>>(row, col, att, h, out, E, N);
}